// CCMCP_GNN_17154099380376
// MI455X (gfx1250) — compile-verified
//
#include <hip/hip_runtime.h>

typedef __attribute__((ext_vector_type(2))) float v2f;
typedef __attribute__((ext_vector_type(8))) float v8f;

// ---------------- degree / normalization ----------------

__global__ void k_init_deg(float* __restrict__ deg, int n) {
  int i = blockIdx.x * blockDim.x + threadIdx.x;
  if (i < n) deg[i] = 1.0f;  // self-loop weight 1
}

__global__ void k_deg_acc(const int* __restrict__ dst, const float* __restrict__ ew,
                          float* __restrict__ deg, int e) {
  int i = blockIdx.x * blockDim.x + threadIdx.x;
  if (i < e) atomicAdd(&deg[dst[i]], ew[i]);
}

__global__ void k_dis(const float* __restrict__ deg, float* __restrict__ dis, int n) {
  int i = blockIdx.x * blockDim.x + threadIdx.x;
  if (i < n) {
    float d = deg[i];
    dis[i] = (d > 0.0f) ? rsqrtf(d) : 0.0f;
  }
}

__global__ void k_edge_norm(const int* __restrict__ src, const int* __restrict__ dst,
                            const float* __restrict__ ew, const float* __restrict__ dis,
                            float* __restrict__ nrm, int e) {
  int i = blockIdx.x * blockDim.x + threadIdx.x;
  if (i < e) nrm[i] = dis[src[i]] * ew[i] * dis[dst[i]];
}

// ---------------- dense GEMMs via fp32 WMMA (16x16x4) ----------------
// One wave computes a 16-row x 16-col output tile.
// A frag (16x4 f32): lane l -> row l&15, K = 2*(l>>4) + {0,1} in {a.x,a.y}
// B frag (4x16 f32): lane l -> col l&15, K = 2*(l>>4) + {0,1} in {b.x,b.y}
// C/D (16x16 f32):   lane l, vgpr v -> row v + 8*(l>>4), col l&15

__global__ void k_gemm1_wmma(const float* __restrict__ x, const float* __restrict__ w1,
                             float* __restrict__ h, int n) {
  int wave = blockIdx.x * (blockDim.x >> 5) + (threadIdx.x >> 5);
  int lane = threadIdx.x & 31;
  int row0 = wave << 4;
  if (row0 >= n) return;                 // wave-uniform: EXEC stays all-ones for WMMA
  int m  = lane & 15;
  int kh = (lane >> 4) << 1;             // 0 or 2
  int row = row0 + m; if (row >= n) row = n - 1;   // clamp (no divergence)

  v8f c = {};
  for (int k0 = 0; k0 < 128; k0 += 4) {
    v2f a, b;
    a.x = x[row * 128 + k0 + kh];
    a.y = x[row * 128 + k0 + kh + 1];
    b.x = w1[(k0 + kh) * 16 + m];
    b.y = w1[(k0 + kh + 1) * 16 + m];
    c = __builtin_amdgcn_wmma_f32_16x16x4_f32(false, a, false, b, (short)0, c,
                                              false, false);
  }
  int half = lane >> 4;
#pragma unroll
  for (int v = 0; v < 8; ++v) {
    int mo = row0 + v + (half << 3);
    if (mo < n) h[mo * 16 + m] = c[v];
  }
}

// Layer 2: A = relu(a1) [Nx16], B = W2 [16x8] zero-padded to 16 cols, out g [Nx8]
__global__ void k_gemm2_wmma(const float* __restrict__ a1, const float* __restrict__ w2,
                             float* __restrict__ g, int n) {
  int wave = blockIdx.x * (blockDim.x >> 5) + (threadIdx.x >> 5);
  int lane = threadIdx.x & 31;
  int row0 = wave << 4;
  if (row0 >= n) return;
  int m  = lane & 15;
  int kh = (lane >> 4) << 1;
  int row = row0 + m; if (row >= n) row = n - 1;

  v8f c = {};
  for (int k0 = 0; k0 < 16; k0 += 4) {
    v2f a, b;
    a.x = fmaxf(a1[row * 16 + k0 + kh], 0.0f);       // fused ReLU
    a.y = fmaxf(a1[row * 16 + k0 + kh + 1], 0.0f);
    b.x = (m < 8) ? w2[(k0 + kh) * 8 + m] : 0.0f;    // pad cols 8..15
    b.y = (m < 8) ? w2[(k0 + kh + 1) * 8 + m] : 0.0f;
    c = __builtin_amdgcn_wmma_f32_16x16x4_f32(false, a, false, b, (short)0, c,
                                              false, false);
  }
  int half = lane >> 4;
#pragma unroll
  for (int v = 0; v < 8; ++v) {
    int mo = row0 + v + (half << 3);
    if (mo < n && m < 8) g[mo * 8 + m] = c[v];
  }
}

// ---------------- self-loop + bias init, edge scatter ----------------

template <int FD>
__global__ void k_self_bias(const float* __restrict__ hin, const float* __restrict__ dis,
                            const float* __restrict__ bias, float* __restrict__ out, int n) {
  int gid = blockIdx.x * blockDim.x + threadIdx.x;
  int i = gid / FD, j = gid % FD;
  if (i < n) {
    float di = dis[i];
    out[gid] = bias[j] + hin[gid] * di * di;   // self-loop: norm = dis[i]*1*dis[i]
  }
}

template <int FD>
__global__ void k_scatter(const int* __restrict__ src, const int* __restrict__ dst,
                          const float* __restrict__ nrm, const float* __restrict__ hin,
                          float* __restrict__ out, int e) {
  int gid = blockIdx.x * blockDim.x + threadIdx.x;
  int ei = gid / FD, j = gid % FD;              // FD consecutive lanes share an edge
  if (ei < e) {
    int s = src[ei], d = dst[ei];
    atomicAdd(&out[d * FD + j], hin[s * FD + j] * nrm[ei]);
  }
}

// ---------------- launch ----------------

extern "C" void kernel_launch(void* const* d_in, const int* in_sizes, int n_in,
                              void* d_out, int out_size, void* d_ws, size_t ws_size,
                              hipStream_t stream) {
  const float* x  = (const float*)d_in[0];
  const int*   ei = (const int*)d_in[1];
  const float* ew = (const float*)d_in[2];
  const float* W1 = (const float*)d_in[3];
  const float* b1 = (const float*)d_in[4];
  const float* W2 = (const float*)d_in[5];
  const float* b2 = (const float*)d_in[6];
  float* out = (float*)d_out;

  const int H = in_sizes[4];              // 16
  const int F = in_sizes[3] / H;          // 128
  const int N = in_sizes[0] / F;          // 100000
  const int E = in_sizes[2];              // 1600000
  (void)n_in; (void)out_size; (void)ws_size;

  const int* src = ei;
  const int* dst = ei + E;

  float* ws  = (float*)d_ws;
  float* deg = ws;                         // N
  float* dis = deg + N;                    // N
  float* nrm = dis + N;                    // E
  float* h   = nrm + E;                    // N*16
  float* a1  = h  + (size_t)N * 16;        // N*16
  float* g   = a1 + (size_t)N * 16;        // N*8

  const int B = 256;                       // 8 waves / block (wave32)
  const int wavesPerBlk = B / 32;
  const int tiles = (N + 15) / 16;
  const int tileBlks = (tiles + wavesPerBlk - 1) / wavesPerBlk;

  // normalization coefficients (shared by both layers)
  k_init_deg <<<(N + B - 1) / B, B, 0, stream>>>(deg, N);
  k_deg_acc  <<<(E + B - 1) / B, B, 0, stream>>>(dst, ew, deg, E);
  k_dis      <<<(N + B - 1) / B, B, 0, stream>>>(deg, dis, N);
  k_edge_norm<<<(E + B - 1) / B, B, 0, stream>>>(src, dst, ew, dis, nrm, E);

  // layer 1: h = x@W1 ; a1 = b1 + self-loop + edge scatter
  k_gemm1_wmma<<<tileBlks, B, 0, stream>>>(x, W1, h, N);
  k_self_bias<16><<<((size_t)N * 16 + B - 1) / B, B, 0, stream>>>(h, dis, b1, a1, N);
  k_scatter<16>  <<<((size_t)E * 16 + B - 1) / B, B, 0, stream>>>(src, dst, nrm, h, a1, E);

  // layer 2: g = relu(a1)@W2 ; out = b2 + self-loop + edge scatter
  k_gemm2_wmma<<<tileBlks, B, 0, stream>>>(a1, W2, g, N);
  k_self_bias<8><<<((size_t)N * 8 + B - 1) / B, B, 0, stream>>>(g, dis, b2, out, N);
  k_scatter<8>  <<<((size_t)E * 8 + B - 1) / B, B, 0, stream>>>(src, dst, nrm, g, out, E);
}